// RoPE_41455024341530
// MI455X (gfx1250) — compile-verified
//
#include <hip/hip_runtime.h>
#include <hip/hip_bf16.h>
#include <math.h>

// RoPE, f32, (B=16, S=8192, D=128). HBM-bandwidth-bound streaming kernel.
//  - 128-bit NT global loads/stores (global_load_b128/global_store_b128, TH=NT)
//  - branch-free Cody-Waite reduction + hardware v_sin_f32/v_cos_f32
//  - wave32: 32 lanes x float4 == one 128-dim head per wave; pos uniform per wave

typedef float v4f __attribute__((ext_vector_type(4)));

#define ROPE_2K   0.41524101186091404f   // 2*log2(10000)/128 : inv_freq[2q] = exp2(-q*2K)
#define ROPE_STEP 0.8659643233600653f    // 10000^(-1/64)     : f(j+1) = f(j)*STEP
#define INV_2PI   0.15915494309189535f
#define PIx2_HI   6.28125f               // 9 mantissa bits -> k*C1 exact for k <= 2^11
#define PIx2_LO   1.9353071795864769e-3f

// sin/cos of a, |a| <= 8192 rad. Reduction error ~2e-7 rad; V_SIN/V_COS take
// revolutions (D = sin(2*pi*S0)), valid for |t| <= 0.5 here.
__device__ __forceinline__ void rope_sincos(float a, float& s, float& c) {
    float k = __builtin_rintf(a * INV_2PI);          // v_rndne_f32, |k| <= 1304
    float r = __builtin_fmaf(-k, PIx2_HI, a);        // exact product, single rounding
    r       = __builtin_fmaf(-k, PIx2_LO, r);        // r in [-pi, pi]
    float t = r * INV_2PI;                           // revolutions in [-0.5, 0.5]
    s = __builtin_amdgcn_sinf(t);                    // v_sin_f32
    c = __builtin_amdgcn_cosf(t);                    // v_cos_f32
}

__global__ __launch_bounds__(256) void rope_f32_kernel(
    const v4f* __restrict__ x4,        // B*S*D/4 float4
    const int* __restrict__ pos,       // B*S int32 token positions
    v4f* __restrict__ out4,            // B*S*D/4 float4
    int total4)                        // number of float4 elements
{
    int g = blockIdx.x * blockDim.x + threadIdx.x;
    if (g >= total4) return;

    int token = g >> 5;                // 32 float4 per 128-dim head
    int q     = g & 31;                // quad q covers pairs j=2q, 2q+1

    float p  = (float)pos[token];      // uniform across the 32 lanes of a head

    float f0 = __builtin_amdgcn_exp2f(-(float)q * ROPE_2K);  // v_exp_f32
    float f1 = f0 * ROPE_STEP;                               // one mul, no 2nd exp2

    float s0, c0, s1, c1;
    rope_sincos(p * f0, s0, c0);
    rope_sincos(p * f1, s1, c1);

    v4f v = __builtin_nontemporal_load(&x4[g]);   // global_load_b128, TH=NT

    v4f o;
    o.x = __builtin_fmaf(v.x, c0, -(v.y * s0));
    o.y = __builtin_fmaf(v.x, s0,   v.y * c0);
    o.z = __builtin_fmaf(v.z, c1, -(v.w * s1));
    o.w = __builtin_fmaf(v.z, s1,   v.w * c1);

    __builtin_nontemporal_store(o, &out4[g]);     // global_store_b128, TH=NT
}

extern "C" void kernel_launch(void* const* d_in, const int* in_sizes, int n_in,
                              void* d_out, int out_size, void* d_ws, size_t ws_size,
                              hipStream_t stream) {
    const v4f* x4   = (const v4f*)d_in[0];   // float32 x, B*S*D elements
    const int* pos  = (const int*)d_in[1];   // integer token_position, B*S elements
    v4f*       out4 = (v4f*)d_out;

    int total4 = in_sizes[0] / 4;            // float4 count
    int block  = 256;                        // 8 wave32 per block
    int grid   = (total4 + block - 1) / block;

    rope_f32_kernel<<<grid, block, 0, stream>>>(x4, pos, out4, total4);
}